// ElementAssembler_38414187496062
// MI455X (gfx1250) — compile-verified
//
#include <hip/hip_runtime.h>

// FEM Q1 Poisson element assembly for MI455X (gfx1250, wave32).
// Phase 1 (VALU f32): per-thread element geometry -> T[e, 2q+i] = w*det*t_i (18 values, padded to 20).
// Phase 2 (WMMA):     fe(16x4) = T(16x20) x G(20x16-pad) as 5 chained V_WMMA_F32_16X16X4_F32.
// Phase 3:            global_atomic_add_f32 scatter into nodal vector.

typedef __attribute__((ext_vector_type(2))) float v2f;
typedef __attribute__((ext_vector_type(8))) float v8f;

#define WAVES_PER_BLOCK 8
#define BLOCK_THREADS (WAVES_PER_BLOCK * 32)

__global__ void zero_kernel(float* __restrict__ out, int n) {
    int i = blockIdx.x * blockDim.x + threadIdx.x;
    if (i < n) out[i] = 0.0f;
}

__global__ __launch_bounds__(BLOCK_THREADS)
void fem_assemble_kernel(const float* __restrict__ points,
                         const int*   __restrict__ elements,
                         const float* __restrict__ u,
                         const float* __restrict__ qw,
                         const float* __restrict__ qp,
                         float*       __restrict__ out,
                         int E) {
    // Per-wave LDS staging: T matrix (32 elems x 20 k) and fe result (32 elems x 4 basis)
    __shared__ float sT[WAVES_PER_BLOCK][32][20];
    __shared__ float sF[WAVES_PER_BLOCK][32][4];
    // Block-shared B operand, pre-packed in the 4x16 f32 B-register layout:
    // sB[ch][lane] = { B_v0, B_v1 } for that lane in chunk ch.
    __shared__ float sB[5][32][2];

    const int wave = threadIdx.x >> 5;
    const int lane = threadIdx.x & 31;
    const int e    = blockIdx.x * BLOCK_THREADS + threadIdx.x;
    const bool valid = (e < E);

    const int nn   = lane & 15;               // A row / B column handled by this lane
    const int koff = (lane < 16) ? 0 : 2;     // local K rows in this lane's registers

    // ------------- Build B = G (20x16, zero-padded 18x4 gradient matrix) once ----------
    // G[2q+i][b] = g[q][b][i]; only wave 0 computes, all waves consume after barrier.
    if (threadIdx.x < 32) {
#pragma unroll
        for (int ch = 0; ch < 5; ++ch) {
            const int k0 = 4 * ch + koff;     // even => k0,k0+1 share quadrature point q
            const int q  = k0 >> 1;
            float bx = 0.0f, by = 0.0f;
            if (q < 9 && nn < 4) {
                const float x = qp[2 * q + 0];
                const float y = qp[2 * q + 1];
                bx = (nn == 0) ? (y - 1.0f) : (nn == 1) ? (1.0f - y) : (nn == 2) ? y : -y;         // i=0
                by = (nn == 0) ? (x - 1.0f) : (nn == 1) ? (-x)       : (nn == 2) ? x : (1.0f - x); // i=1
            }
            sB[ch][lane][0] = bx;
            sB[ch][lane][1] = by;
        }
    }

    // ---------------- Phase 1: per-element geometry (one element per lane) -------------
    float T[20];
#pragma unroll
    for (int k = 0; k < 20; ++k) T[k] = 0.0f;

    int   nidx[4];
    float ex[4], ey[4], uu[4];
    if (valid) {
        const int4 nv = ((const int4*)elements)[e];   // 4 node indices, one b128 load
        nidx[0] = nv.x; nidx[1] = nv.y; nidx[2] = nv.z; nidx[3] = nv.w;
#pragma unroll
        for (int b = 0; b < 4; ++b) {
            const float2 p = ((const float2*)points)[nidx[b]];  // b64 load per node
            ex[b] = p.x;
            ey[b] = p.y;
            uu[b] = u[nidx[b]];
        }
#pragma unroll
        for (int q = 0; q < 9; ++q) {
            const float x = qp[2 * q + 0];
            const float y = qp[2 * q + 1];
            const float w = qw[q];
            // g[q][b][i]: i=0 column depends on y, i=1 column depends on x
            const float g0[4] = { y - 1.0f, 1.0f - y,  y, -y       };
            const float g1[4] = { x - 1.0f, -x,        x, 1.0f - x };
            float a = 0.f, bb = 0.f, c = 0.f, d = 0.f, gu0 = 0.f, gu1 = 0.f;
#pragma unroll
            for (int b = 0; b < 4; ++b) {
                a   += ex[b] * g0[b];   // jac[0][0]
                bb  += ey[b] * g0[b];   // jac[0][1]
                c   += ex[b] * g1[b];   // jac[1][0]
                d   += ey[b] * g1[b];   // jac[1][1]
                gu0 += g0[b] * uu[b];
                gu1 += g1[b] * uu[b];
            }
            const float det = a * d - bb * c;
            const float inv = 1.0f / det;
            const float i00 =  d * inv, i01 = -bb * inv;
            const float i10 = -c * inv, i11 =  a * inv;
            const float s0 = i00 * gu0 + i01 * gu1;
            const float s1 = i10 * gu0 + i11 * gu1;
            const float t0 = i00 * s0 + i10 * s1;
            const float t1 = i01 * s0 + i11 * s1;
            const float wd = w * det;
            T[2 * q + 0] = wd * t0;
            T[2 * q + 1] = wd * t1;
        }
    }

    // Stage T into LDS (row = element-in-wave, col = k)
#pragma unroll
    for (int k = 0; k < 20; ++k) sT[wave][lane][k] = T[k];
    __syncthreads();

    // ---------------- Phase 2: fe = T x G via V_WMMA_F32_16X16X4_F32 -------------------
    // A-matrix (16x4 f32) layout: lanes 0-15 hold row M=lane (v0=K0, v1=K1);
    //                             lanes 16-31 hold row M=lane-16 (v0=K2, v1=K3).
    // B-matrix (4x16 f32) layout mirrors A: lanes 0-15 col N=lane (v0=row0, v1=row1);
    //                                       lanes 16-31 col N=lane-16 (v0=row2, v1=row3).
    v8f acc0 = {};  // elements [wavebase + 0 .. 15]
    v8f acc1 = {};  // elements [wavebase + 16 .. 31]

#pragma unroll
    for (int ch = 0; ch < 5; ++ch) {          // K = 20 in chunks of 4
        const int k0 = 4 * ch + koff;
        const int k1 = k0 + 1;

        v2f Bv;
        Bv.x = sB[ch][lane][0];
        Bv.y = sB[ch][lane][1];

        v2f A0, A1;
        A0.x = sT[wave][nn][k0];       A0.y = sT[wave][nn][k1];
        A1.x = sT[wave][16 + nn][k0];  A1.y = sT[wave][16 + nn][k1];

        acc0 = __builtin_amdgcn_wmma_f32_16x16x4_f32(false, A0, false, Bv,
                                                     (short)0, acc0, false, false);
        acc1 = __builtin_amdgcn_wmma_f32_16x16x4_f32(false, A1, false, Bv,
                                                     (short)0, acc1, false, false);
    }

    // D layout: VGPR v, lane<16 -> (M=v, N=lane); lane>=16 -> (M=v+8, N=lane-16).
    // Only columns N=0..3 (basis index) are meaningful.
    if (nn < 4) {
        const int mbase = (lane < 16) ? 0 : 8;
#pragma unroll
        for (int v = 0; v < 8; ++v) {
            sF[wave][mbase + v][nn]      = acc0[v];
            sF[wave][16 + mbase + v][nn] = acc1[v];
        }
    }
    __syncthreads();

    // ---------------- Phase 3: scatter-add into nodal vector ---------------------------
    if (valid) {
#pragma unroll
        for (int b = 0; b < 4; ++b) {
            atomicAdd(&out[nidx[b]], sF[wave][lane][b]);
        }
    }
}

extern "C" void kernel_launch(void* const* d_in, const int* in_sizes, int n_in,
                              void* d_out, int out_size, void* d_ws, size_t ws_size,
                              hipStream_t stream) {
    const float* points   = (const float*)d_in[0];   // (N, 2)
    const int*   elements = (const int*)  d_in[1];   // (E, 4)
    const float* u        = (const float*)d_in[2];   // (N,)
    const float* quad_w   = (const float*)d_in[3];   // (9,)
    const float* quad_p   = (const float*)d_in[4];   // (9, 2)
    float*       out      = (float*)d_out;

    const int E = in_sizes[1] / 4;
    const int N = out_size;

    // Output is accumulated via atomics: zero it every call (graph-replay safe).
    zero_kernel<<<(N + 255) / 256, 256, 0, stream>>>(out, N);

    const int blocks = (E + BLOCK_THREADS - 1) / BLOCK_THREADS;
    fem_assemble_kernel<<<blocks, BLOCK_THREADS, 0, stream>>>(
        points, elements, u, quad_w, quad_p, out, E);
}